// NaiveSamplerModel_2654289789699
// MI455X (gfx1250) — compile-verified
//
#include <hip/hip_runtime.h>
#include <hip/hip_bf16.h>
#include <stdint.h>

// Shapes fixed by the reference
#define B_   8
#define L_   4096
#define C_   2048
#define M_   4
#define P_   512
#define HW_  64     // H = W = sqrt(L)
#define POOL 64
#define KPP  8      // points per pooled output (P / POOL)
#define ROW_STRIDE 2064   // 2048 + 16-float pad: kills LDS bank conflicts, keeps 16B align

typedef float v8f __attribute__((ext_vector_type(8)));
typedef float v2f __attribute__((ext_vector_type(2)));
typedef int   vi4 __attribute__((vector_size(16)));   // matches builtin param type

#if defined(__gfx1250__) && __has_builtin(__builtin_amdgcn_wmma_f32_16x16x4_f32)
#define USE_WMMA_F32X4 1
#endif

#if defined(__gfx1250__) && __has_builtin(__builtin_amdgcn_global_load_async_to_lds_b128)
typedef __attribute__((address_space(1))) vi4 gvi4;   // global (__device__) int4
typedef __attribute__((address_space(3))) vi4 lvi4;   // LDS (__shared__) int4
#endif

__device__ __forceinline__ void async_cp16(const float* g, float* l) {
#if defined(__gfx1250__) && __has_builtin(__builtin_amdgcn_global_load_async_to_lds_b128)
  // gfx1250 async path: tracked by ASYNCcnt, lands directly in LDS
  __builtin_amdgcn_global_load_async_to_lds_b128(
      (gvi4*)g, (lvi4*)l, 0, 0);
#else
  *(float4*)l = *(const float4*)g;
#endif
}

__device__ __forceinline__ void wait_async0() {
#if defined(__gfx1250__) && __has_builtin(__builtin_amdgcn_s_wait_asynccnt)
  __builtin_amdgcn_s_wait_asynccnt(0);
#elif defined(__gfx1250__)
  asm volatile("s_wait_asynccnt 0" ::: "memory");
#endif
}

// One block per (b, m, o): gathers 8 points x 4 corner rows (4*8KB) into LDS
// asynchronously, blends + pools with fp32 WMMA (A = replicated corner
// weights /8, B = 16 channels of the 4 staged rows, C accumulates over the
// 8 points of the pool chunk).
__global__ __launch_bounds__(256)
void point_sample_pool_kernel(const float* __restrict__ feats,
                              const float* __restrict__ pts,
                              float* __restrict__ out) {
  __shared__ float staged[4 * ROW_STRIDE];   // 4 corner rows x 2048 ch (+pad)
  __shared__ int   s_idx[KPP][4];
  __shared__ float s_wt[KPP][4];

  const int tid = threadIdx.x;
  const int bid = blockIdx.x;
  const int o = bid & (POOL - 1);
  const int m = (bid >> 6) & (M_ - 1);
  const int b = bid >> 8;

  // Threads 0..7 each compute one point's corner indices + bilinear weights
  if (tid < KPP) {
    const int p = o * KPP + tid;
    const float* pp = pts + (((size_t)(b * M_ + m) * P_ + p) * 2);
    const float iy = pp[0] * (float)(HW_ - 1);
    const float ix = pp[1] * (float)(HW_ - 1);
    const float y0f = floorf(iy);
    const float x0f = floorf(ix);
    const float wy = iy - y0f;
    const float wx = ix - x0f;
    int y0 = min(max((int)y0f, 0), HW_ - 1);
    int x0 = min(max((int)x0f, 0), HW_ - 1);
    int y1 = min(y0 + 1, HW_ - 1);
    int x1 = min(x0 + 1, HW_ - 1);
    s_idx[tid][0] = y0 * HW_ + x0;
    s_idx[tid][1] = y0 * HW_ + x1;
    s_idx[tid][2] = y1 * HW_ + x0;
    s_idx[tid][3] = y1 * HW_ + x1;
    const float inv = 1.0f / (float)KPP;   // fold the chunked mean into weights
    s_wt[tid][0] = (1.0f - wx) * (1.0f - wy) * inv;
    s_wt[tid][1] = wx * (1.0f - wy) * inv;
    s_wt[tid][2] = (1.0f - wx) * wy * inv;
    s_wt[tid][3] = wx * wy * inv;
  }
  __syncthreads();

#ifdef USE_WMMA_F32X4
  const int lane = tid & 31;
  const int wv   = tid >> 5;     // wave id 0..7, owns channels [256*wv, 256*wv+256)
  const int half = lane >> 4;    // K-half: lanes 0-15 -> K0/K1, lanes 16-31 -> K2/K3
  const int n    = lane & 15;    // column within 16-channel group
  v8f acc[16];
  const v8f vzero = {0.f, 0.f, 0.f, 0.f, 0.f, 0.f, 0.f, 0.f};
  #pragma unroll
  for (int i = 0; i < 16; ++i) acc[i] = vzero;
#else
  float facc[8];
  #pragma unroll
  for (int j = 0; j < 8; ++j) facc[j] = 0.f;
#endif

  const size_t fb = (size_t)b * L_ * C_;

  for (int p = 0; p < KPP; ++p) {
    // ---- async-stage the 4 corner rows (4 x 2048 floats): 8 B128 per thread
    #pragma unroll
    for (int r = 0; r < 4; ++r) {
      const float* rowp = feats + fb + (size_t)s_idx[p][r] * C_;
      #pragma unroll
      for (int k = 0; k < 2; ++k) {
        const int off16 = tid + 256 * k;               // 0..511 float4 chunks
        async_cp16(rowp + off16 * 4, &staged[r * ROW_STRIDE + off16 * 4]);
      }
    }
    wait_async0();
    __syncthreads();   // all waves' async writes visible

#ifdef USE_WMMA_F32X4
    // A[16x4] fp32, all rows = (w00,w01,w10,w11)/8:
    //   lanes 0-15 hold K0,K1; lanes 16-31 hold K2,K3 (per ISA A 16x4 layout)
    v2f a;
    a.x = s_wt[p][half ? 2 : 0];
    a.y = s_wt[p][half ? 3 : 1];
    const int rb0 = (half ? 2 : 0) * ROW_STRIDE;       // B VGPR0 row
    const int rb1 = rb0 + ROW_STRIDE;                  // B VGPR1 row
    #pragma unroll
    for (int i = 0; i < 16; ++i) {
      const int c = wv * 256 + i * 16 + n;
      v2f bm;
      bm.x = staged[rb0 + c];
      bm.y = staged[rb1 + c];
      acc[i] = __builtin_amdgcn_wmma_f32_16x16x4_f32(
          /*neg_a=*/false, a, /*neg_b=*/false, bm,
          /*c_mod=*/(short)0, acc[i], /*reuse_a=*/false, /*reuse_b=*/false);
    }
#else
    #pragma unroll
    for (int r = 0; r < 4; ++r) {
      const float w = s_wt[p][r];
      #pragma unroll
      for (int j = 0; j < 8; ++j)
        facc[j] += w * staged[r * ROW_STRIDE + tid * 8 + j];
    }
#endif
    __syncthreads();   // protect `staged` before next point overwrites it
  }

  float* op = out + (((size_t)(b * M_ + m) * POOL + o) * C_);
#ifdef USE_WMMA_F32X4
  // Every D row is identical (A rows replicated); lanes 0-15 of each wave
  // emit the 16 channels of each group from accumulator element 0.
  if (lane < 16) {
    #pragma unroll
    for (int i = 0; i < 16; ++i)
      op[wv * 256 + i * 16 + lane] = acc[i][0];
  }
#else
  #pragma unroll
  for (int j = 0; j < 8; ++j) op[tid * 8 + j] = facc[j];
#endif
}

extern "C" void kernel_launch(void* const* d_in, const int* in_sizes, int n_in,
                              void* d_out, int out_size, void* d_ws, size_t ws_size,
                              hipStream_t stream) {
  (void)in_sizes; (void)n_in; (void)out_size; (void)d_ws; (void)ws_size;
  const float* feats = (const float*)d_in[0];   // [8, 4096, 2048] fp32
  const float* pts   = (const float*)d_in[1];   // [8, 4, 512, 2]  fp32
  float* out = (float*)d_out;                   // [8, 4, 64, 2048] fp32

  dim3 grid(B_ * M_ * POOL);   // 2048 blocks: one per (b, m, pooled-row)
  dim3 block(256);             // 8 wave32s
  point_sample_pool_kernel<<<grid, block, 0, stream>>>(feats, pts, out);
}